// VQVAE_28845000360777
// MI455X (gfx1250) — compile-verified
//
#include <hip/hip_runtime.h>

typedef __attribute__((ext_vector_type(2))) float v2f;
typedef __attribute__((ext_vector_type(8))) float v8f;

#define NB   256      // batch
#define NC   64       // code groups
#define NS   4096     // dict size
#define NE   128      // embed dim

// ---------------------------------------------------------------------------
// Kernel 1: d_sq[c][s] = sum_e dict[c][s][e]^2   (262144 rows of 128)
// ---------------------------------------------------------------------------
__global__ __launch_bounds__(256) void vq_dsq_kernel(const float* __restrict__ dict,
                                                     float* __restrict__ dsq) {
    int gid = blockIdx.x * 256 + threadIdx.x;           // [0, NC*NS)
    const float4* p = (const float4*)(dict + (size_t)gid * NE);
    float s = 0.0f;
#pragma unroll 8
    for (int i = 0; i < NE / 4; ++i) {
        float4 q = p[i];
        s += q.x * q.x + q.y * q.y + q.z * q.z + q.w * q.w;
    }
    dsq[gid] = s;
}

// ---------------------------------------------------------------------------
// Kernel 2: fp32 WMMA cross-product + fused argmin over S.
// Grid: (16 b-tiles, 64 c). Block: 128 (4 waves); wave w owns S-chunk
// [w*1024, (w+1)*1024). Per wave: 16x64 tiles, 4 accumulators, K=128.
// score = d_sq[s] - 2*cross  (x^2 term dropped: constant per row for argmin)
// ---------------------------------------------------------------------------
__global__ __launch_bounds__(128) void vq_argmin_kernel(const float* __restrict__ x,
                                                        const float* __restrict__ dict,
                                                        const float* __restrict__ dsq,
                                                        int* __restrict__ idxOut) {
    const int btile = blockIdx.x;        // 0..15  (rows btile*16 .. +15)
    const int c     = blockIdx.y;        // 0..63
    const int tid   = threadIdx.x;
    const int wave  = tid >> 5;
    const int lane  = tid & 31;
    const int col   = lane & 15;         // matrix row (A) / col (B) index
    const int grp   = lane >> 4;         // selects K pair {0,1} vs {2,3}

    __shared__ float ldsA[16 * NE];      // 8 KB A tile, row-major [m][e]
    __shared__ float redMin[4][16];
    __shared__ int   redIdx[4][16];

    // Stage A tile into LDS (uniform, no divergence): x[b0+r][c*128 + e]
    {
        const float* xsrc = x + (size_t)(btile * 16) * (NC * NE) + (size_t)c * NE;
        float4* dst4 = (float4*)ldsA;
#pragma unroll
        for (int i = tid; i < 16 * NE / 4; i += 128) {
            int r = i >> 5, e4 = i & 31;
            dst4[i] = ((const float4*)(xsrc + (size_t)r * (NC * NE)))[e4];
        }
    }
    __syncthreads();

    const float* dictc = dict + (size_t)c * NS * NE;
    const float* dsqc  = dsq + (size_t)c * NS;

    float minv[8];
    int   mini[8];
#pragma unroll
    for (int v = 0; v < 8; ++v) { minv[v] = 3.4e38f; mini[v] = 0; }

    const int sBase = wave * 1024;
    for (int st = 0; st < 1024; st += 64) {
        const int s0 = sBase + st;
        v8f acc[4];
        const v8f zero = {0.f, 0.f, 0.f, 0.f, 0.f, 0.f, 0.f, 0.f};
#pragma unroll
        for (int t = 0; t < 4; ++t) acc[t] = zero;

#pragma unroll 4
        for (int k = 0; k < NE; k += 4) {
            // A 16x4 f32 frag: lane = row M, VGPR pair = K = {2*grp, 2*grp+1}
            v2f a = *(const v2f*)(&ldsA[col * NE + k + 2 * grp]);
#pragma unroll
            for (int t = 0; t < 4; ++t) {
                // B 4x16 f32 frag: lane = col N (dict row s), K pair mirrored
                const float* bp = dictc + (size_t)(s0 + t * 16 + col) * NE + k + 2 * grp;
                v2f bf = *(const v2f*)bp;
                acc[t] = __builtin_amdgcn_wmma_f32_16x16x4_f32(
                    false, a, false, bf, (short)0, acc[t], false, false);
            }
        }

        // Fold tile into running per-row argmin. C/D layout: lane holds col
        // N=lane&15; VGPR v holds row M = 8*(lane>>4) + v.
#pragma unroll
        for (int t = 0; t < 4; ++t) {
            int scol = s0 + t * 16 + col;
            float bias = dsqc[scol];
#pragma unroll
            for (int v = 0; v < 8; ++v) {
                float score = bias - 2.0f * acc[t][v];
                if (score < minv[v]) { minv[v] = score; mini[v] = scol; }
            }
        }
    }

    // Butterfly argmin across the 16 lanes of each half-wave (masks stay
    // inside the 16-lane group). Tie-break: smaller index (matches argmin).
#pragma unroll
    for (int v = 0; v < 8; ++v) {
#pragma unroll
        for (int m = 8; m >= 1; m >>= 1) {
            float ov = __shfl_xor(minv[v], m, 32);
            int   oi = __shfl_xor(mini[v], m, 32);
            if (ov < minv[v] || (ov == minv[v] && oi < mini[v])) {
                minv[v] = ov; mini[v] = oi;
            }
        }
    }

    if (col == 0) {
#pragma unroll
        for (int v = 0; v < 8; ++v) {
            int row = grp * 8 + v;
            redMin[wave][row] = minv[v];
            redIdx[wave][row] = mini[v];
        }
    }
    __syncthreads();

    // Combine the 4 waves' S-chunk winners; 16 threads write idx[b][c].
    if (tid < 16) {
        float mv = redMin[0][tid];
        int   mi = redIdx[0][tid];
#pragma unroll
        for (int w = 1; w < 4; ++w) {
            float ov = redMin[w][tid];
            int   oi = redIdx[w][tid];
            if (ov < mv || (ov == mv && oi < mi)) { mv = ov; mi = oi; }
        }
        idxOut[(btile * 16 + tid) * NC + c] = mi;
    }
}

// ---------------------------------------------------------------------------
// Kernel 3: cw_embed[b][c*128+e] = dict[c][idx[b][c]][e]   (float4 gather)
// ---------------------------------------------------------------------------
__global__ __launch_bounds__(256) void vq_gather_kernel(const int* __restrict__ idx,
                                                        const float* __restrict__ dict,
                                                        float* __restrict__ cw) {
    int t  = blockIdx.x * 256 + threadIdx.x;   // [0, NB*NC*NE/4)
    int e4 = t & 31;                           // NE/4 = 32
    int bc = t >> 5;                           // b*NC + c
    int c  = bc & (NC - 1);
    int code = idx[bc];
    const float4* src = (const float4*)(dict + ((size_t)c * NS + code) * NE);
    ((float4*)cw)[t] = src[e4];
}

// ---------------------------------------------------------------------------
// Kernel 4: one_hot[b][c][s] = (s == idx[b][c])   (streaming float4 fill)
// ---------------------------------------------------------------------------
__global__ __launch_bounds__(256) void vq_onehot_kernel(const int* __restrict__ idx,
                                                        float* __restrict__ oneHot) {
    int t  = blockIdx.x * 256 + threadIdx.x;   // [0, NB*NC*NS/4)
    int s4 = t & 1023;                         // NS/4 = 1024
    int bc = t >> 10;                          // b*NC + c
    int code = idx[bc];
    float4 o = {0.f, 0.f, 0.f, 0.f};
    int j = code - s4 * 4;
    if ((unsigned)j < 4u) ((float*)&o)[j] = 1.0f;
    ((float4*)oneHot)[t] = o;
}

// ---------------------------------------------------------------------------
extern "C" void kernel_launch(void* const* d_in, const int* in_sizes, int n_in,
                              void* d_out, int out_size, void* d_ws, size_t ws_size,
                              hipStream_t stream) {
    const float* x    = (const float*)d_in[0];   // [256, 8192]
    const float* dict = (const float*)d_in[1];   // [64, 4096, 128]

    float* out    = (float*)d_out;
    float* cw     = out;                                  // 256*8192 floats
    float* oneHot = out + (size_t)NB * NC * NE;           // 256*64*4096 floats

    float* dsq = (float*)d_ws;                                        // 1 MB
    int*   idx = (int*)((char*)d_ws + (size_t)NC * NS * sizeof(float)); // 64 KB

    // 1. ||dict row||^2  (also warms dictionary into 192MB L2)
    vq_dsq_kernel<<<(NC * NS) / 256, 256, 0, stream>>>(dict, dsq);

    // 2. WMMA fp32 cross-term + fused argmin
    dim3 g2(NB / 16, NC);
    vq_argmin_kernel<<<g2, 128, 0, stream>>>(x, dict, dsq, idx);

    // 3. gather winning codes
    vq_gather_kernel<<<(NB * NC * NE / 4) / 256, 256, 0, stream>>>(idx, dict, cw);

    // 4. one-hot (bandwidth-bound, 268 MB)
    vq_onehot_kernel<<<(NB * NC * NS / 4) / 256, 256, 0, stream>>>(idx, oneHot);
}